// GIN_89824946028579
// MI455X (gfx1250) — compile-verified
//
#include <hip/hip_runtime.h>

typedef float v2f __attribute__((ext_vector_type(2)));
typedef float v8f __attribute__((ext_vector_type(8)));

#define LDSK 260   // padded LDS row stride: 260 % 64 = 4 -> conflict-free column reads
                   // 260 % 4 = 0 -> float4 stores stay 16B-aligned

// ---------------------------------------------------------------------------
// 16-row-tile WMMA GEMM:  Out = act( A' @ W + bias )
//   A' = optional BN+ReLU transform applied while staging the A tile in LDS.
//   Block = 128 threads (4 waves). Weights pre-packed as float2 rows:
//   Wp[k2 * WstrideP + col] = { W[2*k2][col], W[2*k2+1][col] }, zero-padded cols.
// Template params give a branch-free inner loop (no exec churn, acc pinned):
//   NTT  : total 16-col output tiles (16 for H=256, 1 for OUT<=16)
//   BNA  : apply BatchNorm+ReLU to A while staging
//   RELU : ReLU on output
//   TAIL : column-guard the store (ncols not a multiple of 16)
// ---------------------------------------------------------------------------
template<int NTT, bool BNA, bool RELU, bool TAIL>
__global__ __launch_bounds__(128)
void gemm16_wmma(const float* __restrict__ A, int Astride, int nrowsA,
                 const float2* __restrict__ Wp, int WstrideP,
                 const float* __restrict__ bias,
                 float* __restrict__ Out, int OutStride, int ncols, int nrowsOut,
                 int K,
                 const float* __restrict__ bn_mean, const float* __restrict__ bn_rstd,
                 const float* __restrict__ bn_g, const float* __restrict__ bn_be)
{
    constexpr int NT      = (NTT >= 4) ? 4 : NTT;   // tiles per active wave
    constexpr int NACTIVE = (NTT >= 4) ? NTT / 4 : 1;

    __shared__ __align__(16) float As[16 * LDSK];
    const int rowBase = blockIdx.x * 16;
    const int tid = threadIdx.x;
    const int K4 = K >> 2;

    // ---- Stage A tile (16 x K) into LDS with float4 traffic --------------
    for (int i = tid; i < 16 * K4; i += 128) {
        const int r = i / K4;
        const int c = (i - r * K4) << 2;
        const int row = rowBase + r;
        float4 v;
        if (row < nrowsA) v = *(const float4*)(A + (size_t)row * Astride + c);
        else              v = make_float4(0.f, 0.f, 0.f, 0.f);
        if (BNA) {
            const float4 m  = *(const float4*)(bn_mean + c);
            const float4 rs = *(const float4*)(bn_rstd + c);
            const float4 g  = *(const float4*)(bn_g    + c);
            const float4 be = *(const float4*)(bn_be   + c);
            v.x = fmaxf(g.x * (v.x - m.x) * rs.x + be.x, 0.f);
            v.y = fmaxf(g.y * (v.y - m.y) * rs.y + be.y, 0.f);
            v.z = fmaxf(g.z * (v.z - m.z) * rs.z + be.z, 0.f);
            v.w = fmaxf(g.w * (v.w - m.w) * rs.w + be.w, 0.f);
        }
        *(float4*)&As[r * LDSK + c] = v;
    }
    __syncthreads();

    const int wave = tid >> 5;
    if (wave >= NACTIVE) return;                  // wave-uniform, compile-time bound
    const int lane = tid & 31;
    const int lo = lane & 15;                     // A row (M) / B,C col (N) in tile
    const int hi = lane >> 4;                     // K-half (A/B) / M+8 (C/D) selector
    const int colBase = wave * (NT * 16);

    v8f acc[NT] = {};

    // ---- Branch-free WMMA main loop --------------------------------------
#pragma unroll 4
    for (int kk = 0; kk < K4; ++kk) {
        v2f a = *(const v2f*)&As[lo * LDSK + (kk << 2) + (hi << 1)];
        const float2* wrow = Wp + (size_t)((kk << 1) + hi) * WstrideP;
#pragma unroll
        for (int t = 0; t < NT; ++t) {
            const float2 bw = wrow[colBase + t * 16 + lo];
            v2f b; b.x = bw.x; b.y = bw.y;
            acc[t] = __builtin_amdgcn_wmma_f32_16x16x4_f32(
                false, a, false, b, (short)0, acc[t], false, false);
        }
    }

    // ---- Store: C VGPR r -> row = rowBase + r + 8*hi ---------------------
#pragma unroll
    for (int t = 0; t < NT; ++t) {
        const int col = colBase + t * 16 + lo;
        const float bv = (!TAIL || col < ncols) ? bias[col] : 0.0f;
#pragma unroll
        for (int r = 0; r < 8; ++r) {
            const int row = rowBase + r + 8 * hi;
            float v = acc[t][r] + bv;
            if (RELU) v = fmaxf(v, 0.0f);
            if (row < nrowsOut && (!TAIL || col < ncols))
                Out[(size_t)row * OutStride + col] = v;
        }
    }
}

// ---------------------------------------------------------------------------
// Weight repack: Wp[k2 * NcPad + col] = { W[2k2][col], W[2k2+1][col] }, 0-pad.
// ---------------------------------------------------------------------------
__global__ void repack_w_kernel(const float* __restrict__ W, float2* __restrict__ Wp,
                                int Nc, int NcPad, int total)
{
    int idx = blockIdx.x * blockDim.x + threadIdx.x;
    if (idx >= total) return;
    int k2  = idx / NcPad;
    int col = idx - k2 * NcPad;
    float2 v = make_float2(0.f, 0.f);
    if (col < Nc) {
        v.x = W[(size_t)(2 * k2)     * Nc + col];
        v.y = W[(size_t)(2 * k2 + 1) * Nc + col];
    }
    Wp[idx] = v;
}

// ---------------------------------------------------------------------------
// Helpers
// ---------------------------------------------------------------------------
__global__ void copy4_kernel(float* __restrict__ dst, const float* __restrict__ src,
                             long long count4)
{
    long long i = (long long)blockIdx.x * blockDim.x + threadIdx.x;
    if (i < count4) ((float4*)dst)[i] = ((const float4*)src)[i];
}

__global__ void zero4_kernel(float* __restrict__ dst, long long count4)
{
    long long i = (long long)blockIdx.x * blockDim.x + threadIdx.x;
    if (i < count4) ((float4*)dst)[i] = make_float4(0.f, 0.f, 0.f, 0.f);
}

// t[dst[e], :] += h[src[e], :]   edge-parallel scatter-add; h is L2-resident.
// q4shift = log2(K/4)  (K is 128 or 256 -> shift 5 or 6)
__global__ void edge_scatter_kernel(float* __restrict__ t, const float* __restrict__ h,
                                    const int* __restrict__ src, const int* __restrict__ dst,
                                    long long total, int q4shift, int K)
{
    long long idx = (long long)blockIdx.x * blockDim.x + threadIdx.x;
    if (idx >= total) return;
    const int e = (int)(idx >> q4shift);
    const int c = (int)(idx & ((1 << q4shift) - 1)) << 2;
    const float4 v = *(const float4*)(h + (size_t)src[e] * K + c);
    float* p = t + (size_t)dst[e] * K + c;
    atomicAdd(p + 0, v.x);
    atomicAdd(p + 1, v.y);
    atomicAdd(p + 2, v.z);
    atomicAdd(p + 3, v.w);
}

// gemb[batch[i], :] += h[i, :]
__global__ void pool_kernel(float* __restrict__ gemb, const float* __restrict__ h,
                            const int* __restrict__ batch, long long total)
{
    long long idx = (long long)blockIdx.x * blockDim.x + threadIdx.x;
    if (idx >= total) return;
    const int i = (int)(idx >> 6);
    const int c = (int)(idx & 63) << 2;
    const float4 v = *(const float4*)(h + (size_t)i * 256 + c);
    float* p = gemb + (size_t)batch[i] * 256 + c;
    atomicAdd(p + 0, v.x);
    atomicAdd(p + 1, v.y);
    atomicAdd(p + 2, v.z);
    atomicAdd(p + 3, v.w);
}

// BatchNorm column statistics: deterministic two-level reduction over N rows.
__global__ void bn_partial_kernel(const float* __restrict__ Y, int N, int rpb,
                                  float* __restrict__ ps, float* __restrict__ ps2)
{
    const int c = threadIdx.x;          // 256 columns
    const int b = blockIdx.x;
    const int r0 = b * rpb;
    int r1 = r0 + rpb; if (r1 > N) r1 = N;
    float s = 0.f, s2 = 0.f;
    for (int r = r0; r < r1; ++r) {
        float v = Y[(size_t)r * 256 + c];
        s += v; s2 += v * v;
    }
    ps [b * 256 + c] = s;
    ps2[b * 256 + c] = s2;
}

__global__ void bn_final_kernel(const float* __restrict__ ps, const float* __restrict__ ps2,
                                int nblk, int N,
                                float* __restrict__ mean, float* __restrict__ rstd)
{
    const int c = threadIdx.x;
    float s = 0.f, s2 = 0.f;
    for (int b = 0; b < nblk; ++b) {    // fixed order -> deterministic
        s  += ps [b * 256 + c];
        s2 += ps2[b * 256 + c];
    }
    const float m = s / (float)N;
    float var = fmaxf(s2 / (float)N - m * m, 0.0f);
    mean[c] = m;
    rstd[c] = rsqrtf(var + 1e-5f);
}

// ---------------------------------------------------------------------------
// Orchestration
// ---------------------------------------------------------------------------
extern "C" void kernel_launch(void* const* d_in, const int* in_sizes, int n_in,
                              void* d_out, int out_size, void* d_ws, size_t ws_size,
                              hipStream_t stream)
{
    const float* x     = (const float*)d_in[0];
    const int*   ei    = (const int*)  d_in[1];
    const int*   batch = (const int*)  d_in[2];

    const float* c0_W1 = (const float*)d_in[4];
    const float* c0_b1 = (const float*)d_in[5];
    const float* c0_g  = (const float*)d_in[6];
    const float* c0_be = (const float*)d_in[7];
    const float* c0_W2 = (const float*)d_in[8];
    const float* c0_b2 = (const float*)d_in[9];
    const float* cs_W1 = (const float*)d_in[10];
    const float* cs_b1 = (const float*)d_in[11];
    const float* cs_g  = (const float*)d_in[12];
    const float* cs_be = (const float*)d_in[13];
    const float* cs_W2 = (const float*)d_in[14];
    const float* cs_b2 = (const float*)d_in[15];
    const float* mlp_W1 = (const float*)d_in[16];
    const float* mlp_b1 = (const float*)d_in[17];
    const float* mlp_W2 = (const float*)d_in[18];
    const float* mlp_b2 = (const float*)d_in[19];

    const int N = in_sizes[0] / 128;
    const int E = in_sizes[1] / 2;
    const int G = out_size / 10;
    const int* src = ei;
    const int* dst = ei + E;

    // ---- Workspace layout (fp32) ----
    const size_t NH = (size_t)N * 256;
    float* w    = (float*)d_ws;
    float* t    = w;                 // aggregated features   [N,256] (conv0 uses 128)
    float* Y    = t + NH;            // pre-BN hidden         [N,256]
    float* h    = Y + NH;            // layer output          [N,256]
    float* ps   = h + NH;            // BN partials           [128,256]
    float* ps2  = ps  + 128 * 256;
    float* mean = ps2 + 128 * 256;   // [256]
    float* rstd = mean + 256;        // [256]
    float* gemb = rstd + 256;        // pooled graphs         [512,256]
    float* Yh   = gemb + 512 * 256;  // head hidden           [512,256]
    float2* wpk = (float2*)(Yh + 512 * 256);     // packed weights region
    float2* p_c0W1  = wpk;                       //  64*256
    float2* p_c0W2  = p_c0W1 + 64 * 256;         // 128*256
    float2* p_csW1  = p_c0W2 + 128 * 256;        // 2 x 128*256
    float2* p_csW2  = p_csW1 + 2 * 128 * 256;    // 2 x 128*256
    float2* p_mW1   = p_csW2 + 2 * 128 * 256;    // 128*256
    float2* p_mW2   = p_mW1  + 128 * 256;        // 128*16

    // ---- Repack all GEMM weights (tiny; once per launch) ----
    {
        const int T256 = 128 * 256;
        repack_w_kernel<<<( 64 * 256 + 255) / 256, 256, 0, stream>>>(c0_W1, p_c0W1, 256, 256, 64 * 256);
        repack_w_kernel<<<(T256 + 255) / 256, 256, 0, stream>>>(c0_W2, p_c0W2, 256, 256, T256);
        repack_w_kernel<<<(T256 + 255) / 256, 256, 0, stream>>>(cs_W1,               p_csW1,        256, 256, T256);
        repack_w_kernel<<<(T256 + 255) / 256, 256, 0, stream>>>(cs_W1 + 256 * 256,   p_csW1 + T256, 256, 256, T256);
        repack_w_kernel<<<(T256 + 255) / 256, 256, 0, stream>>>(cs_W2,               p_csW2,        256, 256, T256);
        repack_w_kernel<<<(T256 + 255) / 256, 256, 0, stream>>>(cs_W2 + 256 * 256,   p_csW2 + T256, 256, 256, T256);
        repack_w_kernel<<<(T256 + 255) / 256, 256, 0, stream>>>(mlp_W1, p_mW1, 256, 256, T256);
        repack_w_kernel<<<(128 * 16 + 255) / 256, 256, 0, stream>>>(mlp_W2, p_mW2, 10, 16, 128 * 16);
    }

    const int rowTiles = (N + 15) / 16;
    const int rpb = (N + 127) / 128;

    for (int l = 0; l < 3; ++l) {
        const float* hin = (l == 0) ? x : h;
        const int K1 = (l == 0) ? 128 : 256;
        const float2* W1p = (l == 0) ? p_c0W1 : p_csW1 + (size_t)(l - 1) * 128 * 256;
        const float2* W2p = (l == 0) ? p_c0W2 : p_csW2 + (size_t)(l - 1) * 128 * 256;
        const float* b1 = (l == 0) ? c0_b1 : cs_b1 + (size_t)(l - 1) * 256;
        const float* g  = (l == 0) ? c0_g  : cs_g  + (size_t)(l - 1) * 256;
        const float* be = (l == 0) ? c0_be : cs_be + (size_t)(l - 1) * 256;
        const float* b2 = (l == 0) ? c0_b2 : cs_b2 + (size_t)(l - 1) * 256;

        // t = h_in ; t[dst] += h_in[src]   (GIN aggregation, eps = 0)
        long long c4 = (long long)N * K1 / 4;
        copy4_kernel<<<(int)((c4 + 255) / 256), 256, 0, stream>>>(t, hin, c4);
        const int q4shift = (K1 == 128) ? 5 : 6;
        long long et = (long long)E << q4shift;
        edge_scatter_kernel<<<(int)((et + 255) / 256), 256, 0, stream>>>(t, hin, src, dst, et, q4shift, K1);

        // Y = t @ W1 + b1
        gemm16_wmma<16, false, false, false><<<rowTiles, 128, 0, stream>>>(
            t, K1, N, W1p, 256, b1, Y, 256, 256, N, K1,
            nullptr, nullptr, nullptr, nullptr);

        // BN stats over Y columns
        bn_partial_kernel<<<128, 256, 0, stream>>>(Y, N, rpb, ps, ps2);
        bn_final_kernel<<<1, 256, 0, stream>>>(ps, ps2, 128, N, mean, rstd);

        // h = relu( relu(BN(Y)) @ W2 + b2 )
        gemm16_wmma<16, true, true, false><<<rowTiles, 128, 0, stream>>>(
            Y, 256, N, W2p, 256, b2, h, 256, 256, N, 256,
            mean, rstd, g, be);
    }

    // global_add_pool
    const int Gpad = (G + 15) & ~15;
    zero4_kernel<<<(512 * 256 / 4 + 255) / 256, 256, 0, stream>>>(gemb, 512 * 256 / 4);
    long long pt = (long long)N * 64;
    pool_kernel<<<(int)((pt + 255) / 256), 256, 0, stream>>>(gemb, h, batch, pt);

    // Head MLP: Yh = relu(gemb @ mlp_W1 + b1);  out = Yh @ mlp_W2 + b2
    const int gTiles = Gpad / 16;
    gemm16_wmma<16, false, true, false><<<gTiles, 128, 0, stream>>>(
        gemb, 256, Gpad, p_mW1, 256, mlp_b1, Yh, 256, 256, Gpad, 256,
        nullptr, nullptr, nullptr, nullptr);
    gemm16_wmma<1, false, false, true><<<gTiles, 128, 0, stream>>>(
        Yh, 256, Gpad, p_mW2, 16, mlp_b2, (float*)d_out, 10, 10, G, 256,
        nullptr, nullptr, nullptr, nullptr);
}